// CaLCS_64742337020403
// MI455X (gfx1250) — compile-verified
//
#include <hip/hip_runtime.h>
#include <hip/hip_bf16.h>
#include <math.h>

// Problem constants from the reference setup_inputs(): B=8, r=512, V=32000, c=512.
#define B_  8
#define R_  512
#define V_  32000
#define C_  512
#define DT_ (R_ + C_ - 1)   // 1023 anti-diagonals

typedef float v2f __attribute__((ext_vector_type(2)));
typedef float v8f __attribute__((ext_vector_type(8)));

// gfx1250 async global->LDS path (probe via __has_builtin; falls back cleanly).
#if defined(__AMDGCN__) && __has_builtin(__builtin_amdgcn_global_load_async_to_lds_b32) && \
    __has_builtin(__builtin_amdgcn_s_wait_asynccnt)
#define CALCS_ASYNC 1
#else
#define CALCS_ASYNC 0
#endif

// ---------------------------------------------------------------------------
// Kernel A: one workgroup per (b,i) row of logits.
//   1) online (max, sum-exp) over V=32000 (single 524MB streaming pass total)
//   2) row max via LDS tree; per-thread sums rescaled to the row max
//   3) exact-f32 sum of 256 partials using V_WMMA_F32_16X16X4_F32 (A=data 16x4,
//      B=ones 4x16 -> D[m][n] = rowsum_m; 8 D-regs + shfl_xor(16) = total)
//   4) gather p[j] = exp(x[docs[j]] - M) / S  (row is L2-hot), write row-major
// ---------------------------------------------------------------------------
__global__ __launch_bounds__(256) void calcs_row_kernel(
    const float* __restrict__ batch, const int* __restrict__ docs,
    float* __restrict__ prow)
{
  const int row = blockIdx.x;            // row = b*R_ + i
  const int b   = row / R_;
  const int tid = threadIdx.x;
  const float* __restrict__ x = batch + (size_t)row * V_;

  // --- online max + sum-exp, thread-strided (coalesced) ---
  float m = -3.4e38f, s = 0.0f;
  for (int k = tid; k < V_; k += 256) {
    if (k + 4096 < V_) __builtin_prefetch(x + k + 4096, 0, 0);
    float v  = x[k];
    float mn = fmaxf(m, v);
    s = s * expf(m - mn) + expf(v - mn);
    m = mn;
  }

  __shared__ float red[256];
  __shared__ float rowSinv;

  // --- row max (LDS tree) ---
  red[tid] = m;
  __syncthreads();
  for (int st = 128; st > 0; st >>= 1) {
    if (tid < st) red[tid] = fmaxf(red[tid], red[tid + st]);
    __syncthreads();
  }
  const float M = red[0];
  __syncthreads();                       // everyone has read red[0]

  // --- rescale partial sums to row max, then exact f32 WMMA reduction ---
  red[tid] = s * expf(m - M);
  __syncthreads();
  if (tid < 32) {                        // wave 0 only; EXEC all ones inside
    const int L = tid;
    float a0 = red[L]      + red[L + 64]  + red[L + 128] + red[L + 192];
    float a1 = red[L + 32] + red[L + 96]  + red[L + 160] + red[L + 224];
    // 16x4 f32 A layout: lane m<16 -> A[m][0],A[m][1]; lane m+16 -> A[m][2],A[m][3]
    v2f A;  A.x = a0;  A.y = a1;
    v2f Bones; Bones.x = 1.0f; Bones.y = 1.0f;   // B = ones(4x16), layout-independent
    v8f Cz = {0.f, 0.f, 0.f, 0.f, 0.f, 0.f, 0.f, 0.f};
    v8f D = __builtin_amdgcn_wmma_f32_16x16x4_f32(
        false, A, false, Bones, (short)0, Cz, false, false);
    float z = D[0] + D[1] + D[2] + D[3] + D[4] + D[5] + D[6] + D[7];
    z += __shfl_xor(z, 16, 32);          // combine the two M-halves of D
    if (L == 0) rowSinv = 1.0f / z;
  }
  __syncthreads();
  const float Sinv = rowSinv;

  // --- gather doc-token probabilities, row-major write ---
  const int* __restrict__ drow = docs + b * C_;
  float* __restrict__ pr = prow + (size_t)row * C_;
  for (int j = tid; j < C_; j += 256) {
    int tok = drow[j];
    pr[j] = expf(x[tok] - M) * Sinv;
  }
}

// ---------------------------------------------------------------------------
// Kernel B: re-layout p[b][i][j] -> pd[b][d][i] (diagonal-major, d = i + j),
// zero-filling invalid cells, via a 64x65 LDS tile (coalesced both sides).
// ---------------------------------------------------------------------------
__global__ __launch_bounds__(256) void calcs_diag_kernel(
    const float* __restrict__ prow, float* __restrict__ pdiag)
{
  __shared__ float tile[64][65];
  const int b  = blockIdx.z;
  const int d0 = blockIdx.x * 64;
  const int i0 = blockIdx.y * 64;
  const int tx = threadIdx.x & 63;
  const int ty = threadIdx.x >> 6;       // 0..3

  for (int p = 0; p < 16; ++p) {
    int ii = ty + 4 * p;
    int i  = i0 + ii;
    int j  = d0 + tx - i;                // contiguous in tx for fixed i
    float v = 0.0f;
    if (i < R_ && j >= 0 && j < C_)
      v = prow[((size_t)b * R_ + i) * C_ + j];
    tile[ii][tx] = v;                    // tile[ii][tx] = value at (i0+ii, d0+tx)
  }
  __syncthreads();
  for (int p = 0; p < 16; ++p) {
    int dd = ty + 4 * p;
    int d  = d0 + dd;
    int i  = i0 + tx;
    if (d < DT_ && i < R_)
      pdiag[((size_t)b * DT_ + d) * R_ + i] = tile[tx][dd];
  }
}

// ---------------------------------------------------------------------------
// Kernel C: the wavefront DP. One workgroup per batch, thread i owns row i.
//   D[i,j] = p*(1 + D[i-1,j-1]) + (1-p)*max(D[i-1,j], D[i,j-1])
// 3-bank LDS ring holds neighbor (i-1) values of the last two diagonals;
// own D[i,j-1] stays in a register. One barrier per diagonal step.
// Next diagonal's p is prefetched: async global->LDS double-buffer when the
// toolchain exposes it, register double-buffer otherwise.
// ---------------------------------------------------------------------------
__global__ __launch_bounds__(512) void calcs_dp_kernel(
    const float* __restrict__ pdiag, float* __restrict__ Dout)
{
  const int b = blockIdx.x;
  const int i = threadIdx.x;
  __shared__ float buf[3][R_ + 1];       // slot 0 = boundary zero (i-1 = -1)
  buf[0][i + 1] = 0.f; buf[1][i + 1] = 0.f; buf[2][i + 1] = 0.f;
  if (i == 0) { buf[0][0] = 0.f; buf[1][0] = 0.f; buf[2][0] = 0.f; }

  const float* __restrict__ pdb = pdiag + (size_t)b * DT_ * R_;

#if CALCS_ASYNC
  // Builtin signature (from hipcc diagnostic): (__device__ int*, <lds> int*, imm, imm)
  typedef __attribute__((address_space(1))) int g_i32;
  typedef __attribute__((address_space(3))) int l_i32;
  __shared__ float stage[2][R_];
  // prologue: stage diagonal 0
  __builtin_amdgcn_global_load_async_to_lds_b32(
      (g_i32*)(pdb + i), (l_i32*)&stage[0][i], 0, 0);
#else
  float pv_next = pdb[i];                // register prefetch of diagonal 0
#endif
  __syncthreads();

  float p1own = 0.0f;                    // own D[i, j-1]
  int bw = 0, bs1 = 2, bs2 = 1;          // write bank, prev1 bank, prev2 bank
  for (int d = 0; d < DT_; ++d) {
#if CALCS_ASYNC
    int dn = (d + 1 < DT_) ? d + 1 : d;
    __builtin_amdgcn_global_load_async_to_lds_b32(
        (g_i32*)(pdb + (size_t)dn * R_ + i),
        (l_i32*)&stage[(d + 1) & 1][i], 0, 0);
    __builtin_amdgcn_s_wait_asynccnt(1); // diag d staged; each lane reads its own slot
    float pv = stage[d & 1][i];
#else
    float pv = pv_next;
    int dn = (d + 1 < DT_) ? d + 1 : d;
    pv_next = pdb[(size_t)dn * R_ + i];  // independent of DP state -> latency hidden
#endif
    int   j   = d - i;
    float sh1 = buf[bs1][i];             // D[i-1, j]
    float sh2 = buf[bs2][i];             // D[i-1, j-1]
    float nv  = pv * (1.0f + sh2) + (1.0f - pv) * fmaxf(sh1, p1own);
    nv = (j >= 0 && j < C_) ? nv : 0.0f;
    buf[bw][i + 1] = nv;
    p1own = nv;
    int t = bs2; bs2 = bs1; bs1 = bw; bw = t;   // rotate banks
    __syncthreads();
  }
  if (i == R_ - 1) Dout[b] = fminf(p1own, 100.0f);   // D[r-1, c-1], clamped
}

// ---------------------------------------------------------------------------
// Kernel D: loss = -log( mean_b(clamped D) / c )
// ---------------------------------------------------------------------------
__global__ __launch_bounds__(32) void calcs_loss_kernel(
    const float* __restrict__ Din, float* __restrict__ out)
{
  int t = threadIdx.x;
  float v = (t < B_) ? Din[t] : 0.0f;
  for (int off = 16; off > 0; off >>= 1) v += __shfl_xor(v, off, 32);
  if (t == 0) out[0] = -logf((v / (float)B_) / (float)C_);
}

// ---------------------------------------------------------------------------
extern "C" void kernel_launch(void* const* d_in, const int* in_sizes, int n_in,
                              void* d_out, int out_size, void* d_ws, size_t ws_size,
                              hipStream_t stream) {
  (void)in_sizes; (void)n_in; (void)out_size; (void)ws_size;
  const float* batch = (const float*)d_in[0];   // [B, r, V] fp32
  const int*   docs  = (const int*)d_in[1];     // [B, c] int

  char* ws = (char*)d_ws;
  float* prow  = (float*)ws;                                        // B*R*C  (8.4 MB)
  float* pdiag = (float*)(ws + (size_t)B_ * R_ * C_ * sizeof(float)); // B*DT*R (16.8 MB)
  float* Dres  = (float*)(ws + (size_t)B_ * R_ * C_ * sizeof(float)
                             + (size_t)B_ * DT_ * R_ * sizeof(float)); // B floats

  calcs_row_kernel<<<B_ * R_, 256, 0, stream>>>(batch, docs, prow);
  calcs_diag_kernel<<<dim3((DT_ + 63) / 64, R_ / 64, B_), 256, 0, stream>>>(prow, pdiag);
  calcs_dp_kernel<<<B_, R_, 0, stream>>>(pdiag, Dres);
  calcs_loss_kernel<<<1, 32, 0, stream>>>(Dres, (float*)d_out);
}